// RBGBlock_33638183862865
// MI455X (gfx1250) — compile-verified
//
#include <hip/hip_runtime.h>
#include <hip/hip_bf16.h>

typedef float v2f __attribute__((ext_vector_type(2)));
typedef float v8f __attribute__((ext_vector_type(8)));

#define CIN   256
#define COUT  4
#define SP    32768   // 32*32*32 coarse spatial
#define NBAT  8

// ---------------------------------------------------------------------------
// Stage 1: modulated 1x1x1 conv + prev add.
// Per batch: D(16x32768) = A(16x256) x B(256x32768), A rows 4..15 zero-padded.
// Uses V_WMMA_F32_16X16X4_F32, K-loop step 4.
//   A 16x4 f32 frag: lane m=lane&15; v[0]: K = k+2*half, v[1]: K = k+2*half+1
//   B 4x16  f32 frag: v[j]: K = k+2*j+half, N = lane&15
//   C/D 16x16 f32: VGPR r, lanes 0-15 -> (M=r, N=lane)
// Each wave covers 64 spatial positions (4 N-tiles); each x element is read
// exactly once (bandwidth-optimal for the 256 MB dominant stream).
// ---------------------------------------------------------------------------
__global__ __launch_bounds__(256) void conv1x1_wmma(
    const float* __restrict__ x, const float* __restrict__ S,
    const float* __restrict__ prev, const float* __restrict__ weight,
    float* __restrict__ ystage)
{
  // modulated weight, transposed [k][m], rows m>=4 zero (16 KB LDS)
  __shared__ float wmodT[CIN * 16];

  const int b   = blockIdx.y;
  const int tid = threadIdx.x;

  for (int i = tid; i < CIN * 16; i += 256) {
    const int kk = i >> 4;
    const int m  = i & 15;
    float wv = 0.0f;
    if (m < COUT) wv = weight[m * CIN + kk] * (S[b * CIN + kk] + 1.0f);
    wmodT[i] = wv;
  }
  __syncthreads();

  const int wave = tid >> 5;
  const int lane = tid & 31;
  const int half = lane >> 4;   // 0: lanes 0-15, 1: lanes 16-31
  const int col  = lane & 15;
  const int posBase = blockIdx.x * 512 + wave * 64;

  const float* xb = x + (size_t)b * CIN * SP;

  v8f acc[4] = {};

  for (int k = 0; k < CIN; k += 4) {
    // A fragment (shared across the 4 N-tiles)
    v2f a;
    a[0] = wmodT[(k + 2 * half + 0) * 16 + col];
    a[1] = wmodT[(k + 2 * half + 1) * 16 + col];

#pragma unroll
    for (int t = 0; t < 4; ++t) {
      const float* bp = xb + (size_t)(k + half) * SP + (posBase + t * 16 + col);
      v2f bf;
      bf[0] = bp[0];                  // K = k + half
      bf[1] = bp[2 * (size_t)SP];     // K = k + half + 2
      acc[t] = __builtin_amdgcn_wmma_f32_16x16x4_f32(
          /*neg_a=*/false, a, /*neg_b=*/false, bf,
          /*c_mod=*/(short)0, acc[t], /*reuse_a=*/false, /*reuse_b=*/false);
    }
  }

  // Epilogue: rows M=0..3 live in acc VGPRs 0..3, lanes 0-15.
  const float* pb = prev   + (size_t)b * COUT * SP;
  float*       yb = ystage + (size_t)b * COUT * SP;
  if (half == 0) {
#pragma unroll
    for (int t = 0; t < 4; ++t) {
      const int pos = posBase + t * 16 + col;
#pragma unroll
      for (int r = 0; r < COUT; ++r) {
        yb[(size_t)r * SP + pos] = acc[t][r] + pb[(size_t)r * SP + pos];
      }
    }
  }
}

// ---------------------------------------------------------------------------
// Stage 2: fused trilinear x2 upsample (half-pixel) + depthwise 3x3x3 blur
// (normalized [1,1.7,1], replicate pad). Both are separable linear ops; the
// composition is a 3-tap filter per axis on the COARSE grid with coarse-index
// clamping:  fine f=2m+p:  out = w0(p)*c[m-1] + w1*c[m] + w2(p)*c[m+1]
//   p=0: (1.175, 2.275, 0.25)/3.7     p=1: (0.25, 2.275, 1.175)/3.7
// Each thread produces one 2x2x2 fine block from 27 coarse taps.
// ---------------------------------------------------------------------------
__global__ __launch_bounds__(256) void upsample_blur(
    const float* __restrict__ ystage, float* __restrict__ out)
{
  const int idx = blockIdx.x * 256 + threadIdx.x;  // 8*4*32768 threads
  const int y0 = idx & 31;
  const int x0 = (idx >> 5) & 31;
  const int z0 = (idx >> 10) & 31;
  const int bc = idx >> 15;                        // b*4 + c, 0..31

  const float* src = ystage + (size_t)bc * SP;

  const int zi[3] = { z0 > 0 ? z0 - 1 : 0, z0, z0 < 31 ? z0 + 1 : 31 };
  const int xi[3] = { x0 > 0 ? x0 - 1 : 0, x0, x0 < 31 ? x0 + 1 : 31 };
  const int yi[3] = { y0 > 0 ? y0 - 1 : 0, y0, y0 < 31 ? y0 + 1 : 31 };

  float c[3][3][3];
#pragma unroll
  for (int dz = 0; dz < 3; ++dz)
#pragma unroll
    for (int dx = 0; dx < 3; ++dx)
#pragma unroll
      for (int dy = 0; dy < 3; ++dy)
        c[dz][dx][dy] = src[(zi[dz] * 32 + xi[dx]) * 32 + yi[dy]];

  const float wA = 1.175f / 3.7f;   // outer tap, matching parity
  const float wB = 2.275f / 3.7f;   // center tap
  const float wC = 0.25f  / 3.7f;   // outer tap, opposite parity

  // y axis
  float ay[3][3][2];
#pragma unroll
  for (int dz = 0; dz < 3; ++dz)
#pragma unroll
    for (int dx = 0; dx < 3; ++dx) {
      ay[dz][dx][0] = wA * c[dz][dx][0] + wB * c[dz][dx][1] + wC * c[dz][dx][2];
      ay[dz][dx][1] = wC * c[dz][dx][0] + wB * c[dz][dx][1] + wA * c[dz][dx][2];
    }
  // x axis
  float ax[3][2][2];
#pragma unroll
  for (int dz = 0; dz < 3; ++dz)
#pragma unroll
    for (int py = 0; py < 2; ++py) {
      ax[dz][0][py] = wA * ay[dz][0][py] + wB * ay[dz][1][py] + wC * ay[dz][2][py];
      ax[dz][1][py] = wC * ay[dz][0][py] + wB * ay[dz][1][py] + wA * ay[dz][2][py];
    }
  // z axis + coalesced float2 stores (fy pair is contiguous)
#pragma unroll
  for (int pz = 0; pz < 2; ++pz)
#pragma unroll
    for (int px = 0; px < 2; ++px) {
      float o0, o1;
      if (pz == 0) {
        o0 = wA * ax[0][px][0] + wB * ax[1][px][0] + wC * ax[2][px][0];
        o1 = wA * ax[0][px][1] + wB * ax[1][px][1] + wC * ax[2][px][1];
      } else {
        o0 = wC * ax[0][px][0] + wB * ax[1][px][0] + wA * ax[2][px][0];
        o1 = wC * ax[0][px][1] + wB * ax[1][px][1] + wA * ax[2][px][1];
      }
      const size_t ob = ((size_t)bc * 64 + (2 * z0 + pz)) * 4096
                      + (size_t)(2 * x0 + px) * 64 + 2 * y0;
      *(float2*)(out + ob) = make_float2(o0, o1);
    }
}

extern "C" void kernel_launch(void* const* d_in, const int* in_sizes, int n_in,
                              void* d_out, int out_size, void* d_ws, size_t ws_size,
                              hipStream_t stream) {
  const float* x      = (const float*)d_in[0];   // (8,256,32,32,32)
  const float* S      = (const float*)d_in[1];   // (8,256)
  const float* prev   = (const float*)d_in[2];   // (8,4,32,32,32)
  const float* weight = (const float*)d_in[3];   // (4,256)
  float*       out    = (float*)d_out;           // (8,4,64,64,64)
  float*       ystage = (float*)d_ws;            // 4 MB coarse stage buffer

  // Stage 1: 64 spatial chunks x 8 batches, 256 threads (8 waves x 64 pos)
  conv1x1_wmma<<<dim3(64, NBAT), 256, 0, stream>>>(x, S, prev, weight, ystage);
  // Stage 2: one thread per coarse point (8*4*32768 = 1,048,576)
  upsample_blur<<<dim3(4096), 256, 0, stream>>>(ystage, out);
}